// GCN_8435315770069
// MI455X (gfx1250) — compile-verified
//
#include <hip/hip_runtime.h>
#include <hip/hip_bf16.h>

typedef __attribute__((ext_vector_type(2))) float v2f;
typedef __attribute__((ext_vector_type(8))) float v8f;

#define K_DIM 64  // inner dim is 64 for both layers

__device__ __forceinline__ void atomAddF(float* p, float v) {
    unsafeAtomicAdd(p, v);  // native global_atomic_add_f32 on gfx1250
}

// ---------------- zero fill ----------------
__global__ void k_zero(float* __restrict__ p, int n) {
    int i = blockIdx.x * blockDim.x + threadIdx.x;
    if (i < n) p[i] = 0.0f;
}

// ---------------- degree count (in-degree at dst) ----------------
__global__ void k_degree(const int* __restrict__ edst, float* __restrict__ deg, int E) {
    int e = blockIdx.x * blockDim.x + threadIdx.x;
    if (e < E) atomAddF(&deg[edst[e]], 1.0f);
}

// deg -> deg^{-1/2}, self-loop adds 1 so deg >= 1 always
__global__ void k_dinv(float* __restrict__ deg, int n) {
    int i = blockIdx.x * blockDim.x + threadIdx.x;
    if (i < n) deg[i] = rsqrtf(deg[i] + 1.0f);
}

// ---------------- WMMA GEMM: D[M x NCOLS] = A[M x 64] * B[64 x NCOLS] ----------------
// One wave computes one 16x16 output tile with 16 chained v_wmma_f32_16x16x4_f32.
// fp32 WMMA layouts (ISA 7.12.2):
//   A 16x4 : lanes 0-15 hold M=lane, K={0,1}; lanes 16-31 hold M=lane-16, K={2,3}
//   B 4x16 : lanes 0-15 hold N=lane, K rows {0,1}; lanes 16-31 hold N=lane-16, K rows {2,3}
//   C/D    : VGPR r -> (M=r, N=lane) for lanes 0-15; (M=r+8, N=lane-16) for lanes 16-31
template <int NCOLS>
__global__ void k_gemm_wmma(const float* __restrict__ A, const float* __restrict__ B,
                            float* __restrict__ D, int M) {
    const int lane   = threadIdx.x & 31;
    const int wave   = (blockIdx.x * blockDim.x + threadIdx.x) >> 5;  // wave-uniform
    const int ctiles = NCOLS / 16;
    const int tiles  = (M / 16) * ctiles;
    if (wave >= tiles) return;  // whole-wave exit: EXEC stays all-ones for WMMA

    const int rowBase = (wave / ctiles) * 16;
    const int colBase = (wave % ctiles) * 16;
    const int m  = lane & 15;   // M index for A, N index for B/D
    const int hi = lane >> 4;   // lane half selects K sub-pair
    const int kk = hi * 2;

    const float* __restrict__ arow = A + (size_t)(rowBase + m) * K_DIM;
    v8f c = {0.f, 0.f, 0.f, 0.f, 0.f, 0.f, 0.f, 0.f};

#pragma unroll
    for (int k4 = 0; k4 < K_DIM; k4 += 4) {
        v2f a, b;
        a.x = arow[k4 + kk];
        a.y = arow[k4 + kk + 1];
        b.x = B[(size_t)(k4 + kk)     * NCOLS + colBase + m];
        b.y = B[(size_t)(k4 + kk + 1) * NCOLS + colBase + m];
        c = __builtin_amdgcn_wmma_f32_16x16x4_f32(
                /*neg_a=*/false, a, /*neg_b=*/false, b,
                /*c_mod=*/(short)0, c, /*reuse_a=*/false, /*reuse_b=*/false);
    }

    float* __restrict__ dbase = D + (size_t)rowBase * NCOLS + colBase;
#pragma unroll
    for (int r = 0; r < 8; ++r) {
        dbase[(size_t)(r + hi * 8) * NCOLS + m] = c[r];
    }
}

// ---------------- edge scatter, 64 features, 16 threads/item (float4 each) ----------------
__global__ void k_scatter64(const float* __restrict__ h, const float* __restrict__ dinv,
                            const int* __restrict__ esrc, const int* __restrict__ edst,
                            float* __restrict__ agg, int E, int N) {
    int t    = blockIdx.x * blockDim.x + threadIdx.x;
    int item = t >> 4;
    int f4   = t & 15;
    int total = E + N;  // self-loops appended
    if (item >= total) return;
    int s, d;
    if (item < E) { s = esrc[item]; d = edst[item]; }
    else          { s = d = item - E; }
    float nrm = dinv[s] * dinv[d];
    float4 hv = reinterpret_cast<const float4*>(h + (size_t)s * 64)[f4];
    float* out = agg + (size_t)d * 64 + f4 * 4;
    atomAddF(out + 0, hv.x * nrm);
    atomAddF(out + 1, hv.y * nrm);
    atomAddF(out + 2, hv.z * nrm);
    atomAddF(out + 3, hv.w * nrm);
}

// ---------------- edge scatter, 16 features, 4 threads/item ----------------
__global__ void k_scatter16(const float* __restrict__ h, const float* __restrict__ dinv,
                            const int* __restrict__ esrc, const int* __restrict__ edst,
                            float* __restrict__ agg, int E, int N) {
    int t    = blockIdx.x * blockDim.x + threadIdx.x;
    int item = t >> 2;
    int f4   = t & 3;
    int total = E + N;
    if (item >= total) return;
    int s, d;
    if (item < E) { s = esrc[item]; d = edst[item]; }
    else          { s = d = item - E; }
    float nrm = dinv[s] * dinv[d];
    float4 hv = reinterpret_cast<const float4*>(h + (size_t)s * 16)[f4];
    float* out = agg + (size_t)d * 16 + f4 * 4;
    atomAddF(out + 0, hv.x * nrm);
    atomAddF(out + 1, hv.y * nrm);
    atomAddF(out + 2, hv.z * nrm);
    atomAddF(out + 3, hv.w * nrm);
}

// ---------------- relu(agg + b1), 64-wide bias ----------------
__global__ void k_bias_relu(float* __restrict__ a, const float* __restrict__ b, int n) {
    int i = blockIdx.x * blockDim.x + threadIdx.x;
    if (i < n) {
        float v = a[i] + b[i & 63];
        a[i] = v > 0.0f ? v : 0.0f;
    }
}

// ---------------- sigmoid(out + b2), 16-wide bias, in place on d_out ----------------
__global__ void k_bias_sigmoid(float* __restrict__ a, const float* __restrict__ b, int n) {
    int i = blockIdx.x * blockDim.x + threadIdx.x;
    if (i < n) {
        float v = a[i] + b[i & 15];
        a[i] = 1.0f / (1.0f + __expf(-v));
    }
}

extern "C" void kernel_launch(void* const* d_in, const int* in_sizes, int n_in,
                              void* d_out, int out_size, void* d_ws, size_t ws_size,
                              hipStream_t stream) {
    const float* x  = (const float*)d_in[0];   // [N,64]
    const int*   ei = (const int*)d_in[1];     // [2,E]: row0 = src, row1 = dst
    const float* W1 = (const float*)d_in[2];   // [64,64]
    const float* b1 = (const float*)d_in[3];   // [64]
    const float* W2 = (const float*)d_in[4];   // [64,16]
    const float* b2 = (const float*)d_in[5];   // [16]
    float* out = (float*)d_out;                // [N,16]

    const int N = in_sizes[0] / 64;            // 100000 (multiple of 16)
    const int E = in_sizes[1] / 2;             // 1600000
    const int* esrc = ei;
    const int* edst = ei + E;

    // workspace layout (floats)
    float* ws = (float*)d_ws;
    float* deg  = ws;                                   // [N]      -> becomes dinv
    float* hbuf = ws + 100352;                          // [N*64]   -> reused as [N*16] for layer 2
    float* agg1 = ws + 100352 + (size_t)N * 64;         // [N*64]

    const int B = 256;

    // 1) zero accumulators (deg, agg1, d_out)
    k_zero<<<(N + B - 1) / B, B, 0, stream>>>(deg, N);
    k_zero<<<(N * 64 + B - 1) / B, B, 0, stream>>>(agg1, N * 64);
    k_zero<<<(N * 16 + B - 1) / B, B, 0, stream>>>(out, N * 16);

    // 2) degrees and deg^{-1/2}
    k_degree<<<(E + B - 1) / B, B, 0, stream>>>(edst, deg, E);
    k_dinv<<<(N + B - 1) / B, B, 0, stream>>>(deg, N);

    // 3) h = x @ W1   (WMMA fp32, 4 col-tiles per 16-row strip)
    {
        int tiles = (N / 16) * (64 / 16);
        int blocks = (tiles * 32 + B - 1) / B;
        k_gemm_wmma<64><<<blocks, B, 0, stream>>>(x, W1, hbuf, N);
    }

    // 4) agg1[dst] += h[src] * dinv[src]*dinv[dst]  (+ self-loops)
    {
        long long thr = (long long)(E + N) * 16;
        int blocks = (int)((thr + B - 1) / B);
        k_scatter64<<<blocks, B, 0, stream>>>(hbuf, deg, esrc, edst, agg1, E, N);
    }

    // 5) agg1 = relu(agg1 + b1)
    k_bias_relu<<<(N * 64 + B - 1) / B, B, 0, stream>>>(agg1, b1, N * 64);

    // 6) h2 = agg1 @ W2  (reuse hbuf)
    {
        int tiles = N / 16;
        int blocks = (tiles * 32 + B - 1) / B;
        k_gemm_wmma<16><<<blocks, B, 0, stream>>>(agg1, W2, hbuf, N);
    }

    // 7) out[dst] += h2[src] * norm  (+ self-loops)
    {
        long long thr = (long long)(E + N) * 4;
        int blocks = (int)((thr + B - 1) / B);
        k_scatter16<<<blocks, B, 0, stream>>>(hbuf, deg, esrc, edst, out, E, N);
    }

    // 8) out = sigmoid(out + b2)
    k_bias_sigmoid<<<(N * 16 + B - 1) / B, B, 0, stream>>>(out, b2, N * 16);
}